// SimpleGCN_13752485281891
// MI455X (gfx1250) — compile-verified
//
#include <hip/hip_runtime.h>
#include <math.h>

typedef __attribute__((ext_vector_type(2))) float v2f;
typedef __attribute__((ext_vector_type(8))) float v8f;

static inline int cdiv_i(long long a, int b) { return (int)((a + b - 1) / b); }

// ---------------- degree / normalization ----------------

__global__ void gcn_init_deg(float* deg, int n) {
    int i = blockIdx.x * blockDim.x + threadIdx.x;
    if (i < n) deg[i] = 1.0f;   // self-loop contributes 1 to every node's degree
}

__global__ void gcn_count_deg(const int* __restrict__ dst, float* __restrict__ deg, int E) {
    int e = blockIdx.x * blockDim.x + threadIdx.x;
    if (e < E) atomicAdd(&deg[dst[e]], 1.0f);
}

__global__ void gcn_rsqrt_deg(float* deg, int n) {
    int i = blockIdx.x * blockDim.x + threadIdx.x;
    if (i < n) deg[i] = rsqrtf(deg[i]);   // deg >= 1 always (self-loops)
}

// ---------------- tiny weight transpose: Wt[n*K + k] = W[k*NOUT + n] ----------------

__global__ void gcn_transposeW(const float* __restrict__ W, float* __restrict__ Wt,
                               int K, int NOUT) {
    int idx = blockIdx.x * blockDim.x + threadIdx.x;
    if (idx >= K * NOUT) return;
    int k = idx / NOUT;
    int n = idx % NOUT;
    Wt[n * K + k] = W[idx];
}

// ---------------- dense GEMM via fp32 WMMA 16x16x4 ----------------
// H[M,NOUT] = X[M,K] @ W[K,NOUT], W pre-transposed as Wt[NOUT,K].
// One wave computes one 16x16 output tile; K loop fully unrolled.
// fp32 A 16x4 layout : lanes 0-15 -> M=lane, K=k+{0,1}; lanes 16-31 -> M=lane-16, K=k+{2,3}
// fp32 B 4x16 layout : lanes 0-15 -> N=lane, K=k+{0,1}; lanes 16-31 -> N=lane-16, K=k+{2,3}
// fp32 C/D 16x16     : VGPR r: lanes 0-15 -> (M=r, N=lane); lanes 16-31 -> (M=8+r, N=lane-16)
// Columns >= NOUT load from a clamped (in-bounds) address; D[:,n] depends only on
// B[:,n], so those garbage columns never contaminate stored outputs.
template <int K, int NOUT>
__global__ void gcn_gemm_wmma(const float* __restrict__ X, const float* __restrict__ Wt,
                              float* __restrict__ H, int tilesM) {
    constexpr int TILES_N = (NOUT + 15) / 16;
    const int lane = threadIdx.x & 31;
    const int wave = threadIdx.x >> 5;
    const int tile = blockIdx.x * (blockDim.x >> 5) + wave;
    if (tile >= tilesM * TILES_N) return;      // wave-uniform exit: EXEC stays all-1s

    const int tm    = tile / TILES_N;
    const int tn    = tile % TILES_N;
    const int r16   = lane & 15;
    const int hi    = lane >> 4;               // 0 for lanes 0-15, 1 for lanes 16-31
    const int klo   = hi << 1;                 // K sub-offset 0 or 2
    const int arow  = tm * 16 + r16;
    const int ncol  = tn * 16 + r16;
    const int ncolc = (ncol < NOUT) ? ncol : (NOUT - 1);   // clamp, branch-free

    const float* __restrict__ xrow = X  + (long long)arow * K + klo;
    const float* __restrict__ wrow = Wt + (long long)ncolc * K + klo;

    v8f c = {};
#pragma unroll
    for (int k = 0; k < K; k += 4) {
        v2f a = *(const v2f*)(xrow + k);       // contiguous, 8B aligned
        v2f b = *(const v2f*)(wrow + k);       // contiguous, 8B aligned
        c = __builtin_amdgcn_wmma_f32_16x16x4_f32(
                /*neg_a=*/false, a, /*neg_b=*/false, b,
                /*c_mod=*/(short)0, c, /*reuse_a=*/false, /*reuse_b=*/false);
    }

    if ((NOUT % 16 == 0) || (ncol < NOUT)) {
        const int rbase = tm * 16 + hi * 8;
#pragma unroll
        for (int r = 0; r < 8; ++r)
            H[(long long)(rbase + r) * NOUT + ncol] = c[r];
    }
}

// ---------------- accumulator clear ----------------

__global__ void gcn_zero4(float4* p, long long n4) {
    long long i = (long long)blockIdx.x * blockDim.x + threadIdx.x;
    if (i < n4) p[i] = make_float4(0.f, 0.f, 0.f, 0.f);
}

// ---------------- edge scatter-add (L2-atomic bound) ----------------
// thread t handles one float4 chunk of one edge; consecutive threads cover the
// same edge's contiguous features -> coalesced gather + contiguous atomics.
template <int F>
__global__ void gcn_scatter(const float* __restrict__ h, const int* __restrict__ src,
                            const int* __restrict__ dst, const float* __restrict__ dinv,
                            float* __restrict__ acc, int E) {
    constexpr int CPE = F >> 2;                // float4 chunks per edge
    long long tid = (long long)blockIdx.x * blockDim.x + threadIdx.x;
    long long total = (long long)E * CPE;
    if (tid >= total) return;
    int e = (int)(tid / CPE);
    int c = (int)(tid % CPE);
    int s = src[e], d = dst[e];
    float nrm = dinv[s] * dinv[d];
    float4 hv = ((const float4*)(h + (long long)s * F))[c];
    float* out = acc + (long long)d * F + (c << 2);
    atomicAdd(out + 0, hv.x * nrm);
    atomicAdd(out + 1, hv.y * nrm);
    atomicAdd(out + 2, hv.z * nrm);
    atomicAdd(out + 3, hv.w * nrm);
}

// ---------------- self-loop + bias + ReLU (in place over acc) ----------------

template <int F>
__global__ void gcn_finalize_relu(float* __restrict__ acc, const float* __restrict__ h,
                                  const float* __restrict__ dinv, const float* __restrict__ bias,
                                  int N) {
    long long tid = (long long)blockIdx.x * blockDim.x + threadIdx.x;
    long long total = (long long)N * F;
    if (tid >= total) return;
    int i = (int)(tid / F);
    int f = (int)(tid % F);
    float dv = dinv[i];
    float v = acc[tid] + h[tid] * dv * dv + bias[f];
    acc[tid] = fmaxf(v, 0.0f);
}

// ---------------- last layer: self-loop + bias + log_softmax ----------------

template <int F>
__global__ void gcn_final_lsm(const float* __restrict__ acc, const float* __restrict__ h,
                              const float* __restrict__ dinv, const float* __restrict__ bias,
                              float* __restrict__ out, int N) {
    int i = blockIdx.x * blockDim.x + threadIdx.x;
    if (i >= N) return;
    float dv = dinv[i];
    float d2 = dv * dv;
    float v[F];
    float mx = -INFINITY;
    const long long base = (long long)i * F;
#pragma unroll
    for (int f = 0; f < F; ++f) {
        float t = acc[base + f] + h[base + f] * d2 + bias[f];
        v[f] = t;
        mx = fmaxf(mx, t);
    }
    float s = 0.0f;
#pragma unroll
    for (int f = 0; f < F; ++f) s += expf(v[f] - mx);
    float lse = logf(s) + mx;
#pragma unroll
    for (int f = 0; f < F; ++f) out[base + f] = v[f] - lse;
}

// ---------------- driver ----------------

extern "C" void kernel_launch(void* const* d_in, const int* in_sizes, int n_in,
                              void* d_out, int out_size, void* d_ws, size_t ws_size,
                              hipStream_t stream) {
    const float* x  = (const float*)d_in[0];
    const int*   ei = (const int*)d_in[1];
    const float* W1 = (const float*)d_in[2];
    const float* b1 = (const float*)d_in[3];
    const float* W2 = (const float*)d_in[4];
    const float* b2 = (const float*)d_in[5];
    const float* W3 = (const float*)d_in[6];
    const float* b3 = (const float*)d_in[7];
    float* out = (float*)d_out;

    const int N = in_sizes[0] / 128;     // 100000 nodes (multiple of 16)
    const int E = in_sizes[1] / 2;       // 3.2M edges
    const int* src = ei;
    const int* dst = ei + E;

    // workspace layout (floats): dinv[N] | H[N*64] | A[N*64] | Wt[128*64]  (~52 MB)
    float* ws   = (float*)d_ws;
    float* dinv = ws;
    float* H    = ws + N;
    float* A    = H + (size_t)N * 64;
    float* Wt   = A + (size_t)N * 64;

    const int B = 256;
    const int tilesM = N >> 4;

    // --- normalization coefficients ---
    gcn_init_deg<<<cdiv_i(N, B), B, 0, stream>>>(dinv, N);
    gcn_count_deg<<<cdiv_i(E, B), B, 0, stream>>>(dst, dinv, E);
    gcn_rsqrt_deg<<<cdiv_i(N, B), B, 0, stream>>>(dinv, N);

    // --- layer 1: 128 -> 64 ---
    {
        constexpr int K = 128, F = 64, TN = 4;
        gcn_transposeW<<<cdiv_i(K * F, B), B, 0, stream>>>(W1, Wt, K, F);
        gcn_gemm_wmma<K, F><<<cdiv_i((long long)tilesM * TN, 8), B, 0, stream>>>(x, Wt, H, tilesM);
        gcn_zero4<<<cdiv_i((long long)N * F / 4, B), B, 0, stream>>>((float4*)A, (long long)N * F / 4);
        gcn_scatter<F><<<cdiv_i((long long)E * (F / 4), B), B, 0, stream>>>(H, src, dst, dinv, A, E);
        gcn_finalize_relu<F><<<cdiv_i((long long)N * F, B), B, 0, stream>>>(A, H, dinv, b1, N);
    }

    // --- layer 2: 64 -> 32 ---
    {
        constexpr int K = 64, F = 32, TN = 2;
        gcn_transposeW<<<cdiv_i(K * F, B), B, 0, stream>>>(W2, Wt, K, F);
        gcn_gemm_wmma<K, F><<<cdiv_i((long long)tilesM * TN, 8), B, 0, stream>>>(A, Wt, H, tilesM);
        gcn_zero4<<<cdiv_i((long long)N * F / 4, B), B, 0, stream>>>((float4*)A, (long long)N * F / 4);
        gcn_scatter<F><<<cdiv_i((long long)E * (F / 4), B), B, 0, stream>>>(H, src, dst, dinv, A, E);
        gcn_finalize_relu<F><<<cdiv_i((long long)N * F, B), B, 0, stream>>>(A, H, dinv, b2, N);
    }

    // --- layer 3: 32 -> 40, then log_softmax ---
    {
        constexpr int K = 32, F = 40, TN = 3;
        gcn_transposeW<<<cdiv_i(K * F, B), B, 0, stream>>>(W3, Wt, K, F);
        gcn_gemm_wmma<K, F><<<cdiv_i((long long)tilesM * TN, 8), B, 0, stream>>>(A, Wt, H, tilesM);
        gcn_zero4<<<cdiv_i((long long)N * F / 4, B), B, 0, stream>>>((float4*)A, (long long)N * F / 4);
        gcn_scatter<F><<<cdiv_i((long long)E * (F / 4), B), B, 0, stream>>>(H, src, dst, dinv, A, E);
        gcn_final_lsm<F><<<cdiv_i(N, B), B, 0, stream>>>(A, H, dinv, b3, out, N);
    }
}